// SignatureBlock_25967372271865
// MI455X (gfx1250) — compile-verified
//
#include <hip/hip_runtime.h>

typedef float v2f __attribute__((ext_vector_type(2)));
typedef float v4f __attribute__((ext_vector_type(4)));
typedef float v8f __attribute__((ext_vector_type(8)));

#define T_LEN      4096
#define DIMS       8
#define SIGC       584      // 8 + 64 + 512
#define LDS_STRIDE 596      // padded row stride (floats); 596 mod 64 = 20 -> conflict-free A loads
#define MTILE      32       // positions per block
#define KITERS     146      // 584 / 4 (K of each f32 WMMA is 4)
#define NPOS       (8 * T_LEN)

// ---------------------------------------------------------------------------
// Repack w_proj (584 x 128, row-major) into WMMA-B fragment order so the GEMM
// loop does one fully-coalesced global_load_b64 per lane per K-step.
// wsB[((ntile*146 + kiter)*32 + lane)*2 + j] =
//      w[(kiter*4 + 2*(lane>>4) + j)*128 + ntile*16 + (lane&15)]
// ---------------------------------------------------------------------------
__global__ void repack_b_kernel(const float* __restrict__ w,
                                float* __restrict__ wsB) {
    int j = blockIdx.x * blockDim.x + threadIdx.x;
    if (j >= 8 * KITERS * 32) return;
    int lane  = j & 31;
    int kiter = (j >> 5) % KITERS;
    int ntile = j / (32 * KITERS);
    int k0 = kiter * 4 + 2 * (lane >> 4);
    int n  = ntile * 16 + (lane & 15);
    wsB[2 * j + 0] = w[(size_t)k0 * 128 + n];
    wsB[2 * j + 1] = w[(size_t)(k0 + 1) * 128 + n];
}

// ---------------------------------------------------------------------------
// Fused: depth-3 rolling-window signature (Chen recursion, 8 lanes/position)
// -> LDS A-tile -> f32 WMMA projection -> out.
// ---------------------------------------------------------------------------
__global__ __launch_bounds__(256) void sig_gemm_kernel(
    const float* __restrict__ x,      // (8, 4096, 8)
    const float* __restrict__ wsB,    // packed B fragments
    const float* __restrict__ bias,   // (128,)
    float* __restrict__ out)          // (8, 4096, 128)
{
    extern __shared__ float lds[];    // MTILE * LDS_STRIDE floats

    const int tid = threadIdx.x;
    const int row = tid >> 3;         // 0..31 : position within block
    const int i   = tid & 7;          // owned leading tensor index
    const int pos = blockIdx.x * MTILE + row;
    const int b   = pos >> 12;        // / 4096
    const int t   = pos & (T_LEN - 1);

    // -------- Phase 1: signature slice owned by this lane --------
    float s1 = 0.0f;
    float s2r[8];
    float s3r[64];
#pragma unroll
    for (int j = 0; j < 8; ++j)  s2r[j] = 0.0f;
#pragma unroll
    for (int j = 0; j < 64; ++j) s3r[j] = 0.0f;

    const float* xb = x + (size_t)b * T_LEN * DIMS;
    int t0 = t - 7; if (t0 < 0) t0 = 0;

    float pv[8];
    *(v4f*)&pv[0] = *(const v4f*)(xb + (size_t)t0 * DIMS);
    *(v4f*)&pv[4] = *(const v4f*)(xb + (size_t)t0 * DIMS + 4);
    float pvi = xb[(size_t)t0 * DIMS + i];   // own component (dynamic idx via load)

#pragma unroll
    for (int s = 1; s < 8; ++s) {
        int ts = t - 7 + s; if (ts < 0) ts = 0;
        float pn[8];
        *(v4f*)&pn[0] = *(const v4f*)(xb + (size_t)ts * DIMS);
        *(v4f*)&pn[4] = *(const v4f*)(xb + (size_t)ts * DIMS + 4);
        float pni = xb[(size_t)ts * DIMS + i];

        float d[8];
#pragma unroll
        for (int j = 0; j < 8; ++j) d[j] = pn[j] - pv[j];
        const float di = pni - pvi;

        // Chen step (all with OLD s1/s2):
        //   s3[jk] += (s2[j] + (0.5*s1 + di/6)*d_j) * d_k
        //   s2[j]  += (s1 + 0.5*di) * d_j
        //   s1     += di
        const float c3 = 0.5f * s1 + di * (1.0f / 6.0f);
        const float c2 = s1 + 0.5f * di;
#pragma unroll
        for (int j = 0; j < 8; ++j) {
            const float aj = s2r[j] + c3 * d[j];
#pragma unroll
            for (int k = 0; k < 8; ++k)
                s3r[8 * j + k] = fmaf(aj, d[k], s3r[8 * j + k]);
            s2r[j] = fmaf(c2, d[j], s2r[j]);
        }
        s1 += di;

#pragma unroll
        for (int j = 0; j < 8; ++j) pv[j] = pn[j];
        pvi = pni;
    }

    // -------- Store signature slice into the LDS A-tile --------
    float* lrow = lds + row * LDS_STRIDE;
    lrow[i] = s1;                                   // channels [0,8)
#pragma unroll
    for (int j = 0; j < 8; j += 4)                  // channels [8+8i, 8+8i+8)
        *(v4f*)&lrow[8 + 8 * i + j] = *(const v4f*)&s2r[j];
#pragma unroll
    for (int j = 0; j < 64; j += 4)                 // channels [72+64i, 72+64i+64)
        *(v4f*)&lrow[72 + 64 * i + j] = *(const v4f*)&s3r[j];

    __syncthreads();

    // -------- Phase 2: C(32x128) += A(32x584) * B(584x128) via f32 WMMA ----
    const int lane = tid & 31;
    const int wv   = tid >> 5;          // wave id == N-tile (0..7)
    const int mloc = lane & 15;
    const int koff = (lane >> 4) * 2;   // A fragment K offset within 4

    v8f c0 = {0.f, 0.f, 0.f, 0.f, 0.f, 0.f, 0.f, 0.f};
    v8f c1 = {0.f, 0.f, 0.f, 0.f, 0.f, 0.f, 0.f, 0.f};

    const float* bbase  = wsB + ((size_t)wv * KITERS * 32 + lane) * 2;
    const float* a0base = lds + (size_t)mloc * LDS_STRIDE;          // M-tile 0
    const float* a1base = lds + (size_t)(16 + mloc) * LDS_STRIDE;   // M-tile 1

    for (int kk = 0; kk < KITERS; ++kk) {
        v2f bf = *(const v2f*)(bbase + (size_t)kk * 64);
        v2f a0 = *(const v2f*)(a0base + kk * 4 + koff);
        v2f a1 = *(const v2f*)(a1base + kk * 4 + koff);
        c0 = __builtin_amdgcn_wmma_f32_16x16x4_f32(
                 false, a0, false, bf, (short)0, c0, false, false);
        c1 = __builtin_amdgcn_wmma_f32_16x16x4_f32(
                 false, a1, false, bf, (short)0, c1, false, false);
    }

    // -------- Bias + store --------
    const int n  = wv * 16 + mloc;
    const float bi = bias[n];
    const int mhalf = (lane >> 4) * 8;
    const int posbase = blockIdx.x * MTILE;
#pragma unroll
    for (int r = 0; r < 8; ++r) {
        const int m = r + mhalf;
        out[(size_t)(posbase + m) * 128 + n]      = c0[r] + bi;
        out[(size_t)(posbase + 16 + m) * 128 + n] = c1[r] + bi;
    }
}

extern "C" void kernel_launch(void* const* d_in, const int* in_sizes, int n_in,
                              void* d_out, int out_size, void* d_ws, size_t ws_size,
                              hipStream_t stream) {
    const float* x  = (const float*)d_in[0];   // (8,4096,8)
    const float* w  = (const float*)d_in[1];   // (584,128)
    const float* bp = (const float*)d_in[2];   // (128,)
    float* out = (float*)d_out;                // (8,4096,128)
    float* wsB = (float*)d_ws;                 // 74752 floats (299 KB)

    const int nrepack = 8 * KITERS * 32;       // 37376 threads
    repack_b_kernel<<<(nrepack + 255) / 256, 256, 0, stream>>>(w, wsB);

    const int nblocks = NPOS / MTILE;          // 1024
    const size_t shmem = (size_t)MTILE * LDS_STRIDE * sizeof(float); // 76288 B
    sig_gemm_kernel<<<nblocks, 256, shmem, stream>>>(x, wsB, bp, out);
}